// observer_interp2d_11175504904483
// MI455X (gfx1250) — compile-verified
//
#include <hip/hip_runtime.h>

// Problem geometry (fixed by the reference setup).
constexpr int B  = 16;
constexpr int C  = 16;
constexpr int H  = 512;
constexpr int W  = 512;
constexpr int N  = 16384;
constexpr int HW = H * W;

constexpr int THREADS          = 256;          // 8 wave32 waves per block
constexpr int BLOCKS_PER_BATCH = N / THREADS;  // 64, exact
constexpr int CHUNK            = 8;            // channels per load burst

// One thread per (batch, point). Channel loop unrolled in chunks of 8; the
// compiler merges each row-pair of adjacent b32 gathers into global_load_b64
// and clauses them, giving 16 b64 gathers in flight per burst.
__global__ __launch_bounds__(THREADS)
void observer_interp2d_kernel(const float* __restrict__ R,
                              const float* __restrict__ XY,
                              float* __restrict__ out)
{
    // b is uniform per block -> batch bases live in SGPRs.
    const int b = blockIdx.x / BLOCKS_PER_BATCH;
    const int n = (blockIdx.x % BLOCKS_PER_BATCH) * THREADS + threadIdx.x;
    if (b >= B) return;

    // Coalesced, non-temporal point loads (read exactly once): XY is [B,2,N].
    const float* __restrict__ xy = XY + (size_t)b * 2 * N;
    float x = __builtin_nontemporal_load(xy + n)     * (float)(W - 1);
    float y = __builtin_nontemporal_load(xy + N + n) * (float)(H - 1);
    x = fminf(fmaxf(x, 0.0f), (float)(W - 1));
    y = fminf(fmaxf(y, 0.0f), (float)(H - 1));

    // Shifted-window clamp: x0 in [0, W-2], weight wx = x - x0 in [0, 1].
    // At x == W-1 this yields wx == 1.0 on texel W-1 — identical to the
    // reference's (x1 = min(x0+1, W-1), wx = 0) formulation, but the 2x2
    // footprint is always fully in-bounds.
    int x0 = min((int)floorf(x), W - 2);
    int y0 = min((int)floorf(y), H - 2);
    const float wx = x - (float)x0;
    const float wy = y - (float)y0;
    const float w00 = (1.0f - wx) * (1.0f - wy);
    const float w01 = wx * (1.0f - wy);
    const float w10 = (1.0f - wx) * wy;
    const float w11 = wx * wy;

    const float* __restrict__ base =
        R + (size_t)b * C * HW + (size_t)(y0 * W + x0);
    float* __restrict__ o = out + (size_t)b * C * N + n;

    // Warm translation/L2 for the second chunk's plane region while the first
    // chunk's gathers are in flight (gfx1250 global_prefetch_b8).
    __builtin_prefetch(base + (size_t)CHUNK * HW, 0, 0);

#pragma unroll
    for (int cb = 0; cb < C; cb += CHUNK) {
        float v00[CHUNK], v01[CHUNK], v10[CHUNK], v11[CHUNK];
        // Burst-issue 4*CHUNK = 32 independent gathers (R stays RT: it is the
        // only data with L2 reuse — keep it resident).
#pragma unroll
        for (int c = 0; c < CHUNK; ++c) {
            const float* __restrict__ p = base + (size_t)(cb + c) * HW;
            v00[c] = p[0];
            v01[c] = p[1];
            v10[c] = p[W];
            v11[c] = p[W + 1];
        }
        // Blend + coalesced non-temporal b32 stores (written once, never
        // re-read -> don't let the 16 MB output evict R planes from L2).
#pragma unroll
        for (int c = 0; c < CHUNK; ++c) {
            float r = v00[c] * w00;
            r = fmaf(v01[c], w01, r);
            r = fmaf(v10[c], w10, r);
            r = fmaf(v11[c], w11, r);
            __builtin_nontemporal_store(r, o + (size_t)(cb + c) * N);
        }
    }
}

extern "C" void kernel_launch(void* const* d_in, const int* in_sizes, int n_in,
                              void* d_out, int out_size, void* d_ws, size_t ws_size,
                              hipStream_t stream)
{
    (void)in_sizes; (void)n_in; (void)d_ws; (void)ws_size; (void)out_size;

    const float* R  = (const float*)d_in[0];   // [16,16,512,512] f32
    const float* XY = (const float*)d_in[1];   // [16,2,16384]    f32
    float* out      = (float*)d_out;           // [16,16,16384]   f32

    dim3 grid(B * BLOCKS_PER_BATCH);           // 1024 blocks
    dim3 block(THREADS);                       // 256 threads = 8 waves
    observer_interp2d_kernel<<<grid, block, 0, stream>>>(R, XY, out);
}